// DASTNet_62594853372094
// MI455X (gfx1250) — compile-verified
//
#include <hip/hip_runtime.h>
#include <math.h>

// ---------------------------------------------------------------------------
// DASTNet forward for MI455X (gfx1250, wave32).
// All matmuls use V_WMMA_F32_16X16X4_F32 (fp32 in/out, K=4 per instruction).
// ---------------------------------------------------------------------------

constexpr int N_NODES = 2048;
constexpr int IN_D    = 128;
constexpr int H       = 64;
constexpr int SEQL    = 12;
constexpr float BNEPS = 1e-5f;

typedef __attribute__((ext_vector_type(2))) float v2f;
typedef __attribute__((ext_vector_type(8))) float v8f;

// One wave computes a 16x16 f32 tile: c += A(16xK, row-major lda) @ B(Kx16, row-major ldb)
// A frag: lane ml=lane&15 holds M=ml; vgpr pair holds K = kh, kh+1 with kh = 2*(lane>=16).
// B frag: same K pairing, N = ml.  C/D: vgpr r holds M = r + 8*(lane>=16), N = ml.
__device__ __forceinline__ v8f wmma_tile(const float* __restrict__ As, int lda,
                                         const float* __restrict__ Bs, int ldb,
                                         int K, v8f c) {
  const int lane = threadIdx.x & 31;
  const int ml = lane & 15;
  const int kh = (lane >> 4) << 1;
  #pragma unroll
  for (int k = 0; k < K; k += 4) {
    v2f a, b;
    a.x = As[ml * lda + k + kh];
    a.y = As[ml * lda + k + kh + 1];
    b.x = Bs[(k + kh) * ldb + ml];
    b.y = Bs[(k + kh + 1) * ldb + ml];
    c = __builtin_amdgcn_wmma_f32_16x16x4_f32(false, a, false, b, (short)0, c,
                                              false, false);
  }
  return c;
}

// ---------------------------------------------------------------------------
// Extractor kernels
// ---------------------------------------------------------------------------

// xh = h @ ew1^T + eb1   [2048,128]@[128,64]
__global__ __launch_bounds__(256) void k_xh(const float* __restrict__ h,
                                            const float* __restrict__ ew1,
                                            const float* __restrict__ eb1,
                                            float* __restrict__ xh) {
  extern __shared__ float smem[];
  float* sH = smem;          // 64 x 128
  float* sE = smem + 8192;   // 128 x 64 : sE[k][j] = ew1[j][k]
  const int tid = threadIdx.x, lane = tid & 31, wave = tid >> 5;
  const int ml = lane & 15, hi8 = (lane >> 4) << 3;
  const int n0 = blockIdx.x * 64;
  for (int i = tid; i < 64 * 128; i += 256)
    sH[i] = h[(n0 + (i >> 7)) * IN_D + (i & 127)];
  for (int i = tid; i < 128 * 64; i += 256)
    sE[i] = ew1[(i & 63) * IN_D + (i >> 6)];
  __syncthreads();
  for (int tt = wave; tt < 16; tt += 8) {
    const int m0 = (tt >> 2) << 4, c0 = (tt & 3) << 4;
    v8f acc;
    const float bias = eb1[c0 + ml];
    #pragma unroll
    for (int r = 0; r < 8; ++r) acc[r] = bias;
    acc = wmma_tile(&sH[m0 * IN_D], IN_D, &sE[c0], 64, IN_D, acc);
    #pragma unroll
    for (int r = 0; r < 8; ++r)
      xh[(n0 + m0 + r + hi8) * H + c0 + ml] = acc[r];
  }
}

// deg[n] = max-guarded row-sum of adj
__global__ __launch_bounds__(256) void k_deg(const float* __restrict__ adj,
                                             float* __restrict__ deg) {
  const int lane = threadIdx.x & 31, wave = threadIdx.x >> 5;
  const int row = blockIdx.x * 8 + wave;
  float s = 0.f;
  for (int c = lane; c < N_NODES; c += 32) s += adj[(size_t)row * N_NODES + c];
  #pragma unroll
  for (int off = 16; off; off >>= 1) s += __shfl_xor(s, off, 32);
  if (lane == 0) deg[row] = (s < 1e-6f) ? 1.f : s;
}

// x2 = (adj @ xh)/deg + eps1*xh     [2048,2048]@[2048,64]
__global__ __launch_bounds__(256) void k_pool(const float* __restrict__ adj,
                                              const float* __restrict__ xh,
                                              const float* __restrict__ deg,
                                              const float* __restrict__ eps1,
                                              float* __restrict__ x2) {
  extern __shared__ float smem[];
  float* sA = smem;         // 64 x 64 adj tile
  float* sB = smem + 4096;  // 64 x 64 xh tile
  const int tid = threadIdx.x, lane = tid & 31, wave = tid >> 5;
  const int ml = lane & 15, hi8 = (lane >> 4) << 3;
  const int n0 = blockIdx.x * 64;
  const int t0 = wave, t1 = wave + 8;
  const int m00 = (t0 >> 2) << 4, c00 = (t0 & 3) << 4;
  const int m01 = (t1 >> 2) << 4, c01 = (t1 & 3) << 4;
  v8f acc0, acc1;
  #pragma unroll
  for (int r = 0; r < 8; ++r) { acc0[r] = 0.f; acc1[r] = 0.f; }
  for (int kb = 0; kb < N_NODES; kb += 64) {
    for (int i = tid; i < 64 * 64; i += 256) {
      const int r = i >> 6, c = i & 63;
      sA[i] = adj[(size_t)(n0 + r) * N_NODES + kb + c];
      sB[i] = xh[(kb + r) * H + c];
    }
    if (kb + 64 < N_NODES)  // warm next adj tile (global_prefetch_b8)
      __builtin_prefetch(&adj[(size_t)(n0 + (tid & 63)) * N_NODES + kb + 64 +
                              ((tid >> 6) << 4)], 0, 1);
    __syncthreads();
    acc0 = wmma_tile(&sA[m00 * 64], 64, &sB[c00], 64, 64, acc0);
    acc1 = wmma_tile(&sA[m01 * 64], 64, &sB[c01], 64, 64, acc1);
    __syncthreads();
  }
  const float ep = eps1[0];
  #pragma unroll
  for (int r = 0; r < 8; ++r) {
    int g = n0 + m00 + r + hi8;
    x2[g * H + c00 + ml] = acc0[r] / deg[g] + ep * xh[g * H + c00 + ml];
    g = n0 + m01 + r + hi8;
    x2[g * H + c01 + ml] = acc1[r] / deg[g] + ep * xh[g * H + c01 + ml];
  }
}

// column mean / biased var over 2048 rows -> mv[0:64]=mean, mv[64:128]=var
__global__ __launch_bounds__(256) void k_stats(const float* __restrict__ x2,
                                               float* __restrict__ mv) {
  __shared__ float sS[256], sQ[256];
  const int col = threadIdx.x & 63, g = threadIdx.x >> 6;
  float s = 0.f, q = 0.f;
  for (int r = g * 512; r < (g + 1) * 512; ++r) {
    const float v = x2[r * H + col];
    s += v; q += v * v;
  }
  sS[threadIdx.x] = s; sQ[threadIdx.x] = q;
  __syncthreads();
  if (threadIdx.x < 64) {
    float ts = 0.f, tq = 0.f;
    #pragma unroll
    for (int i = 0; i < 4; ++i) { ts += sS[i * 64 + threadIdx.x]; tq += sQ[i * 64 + threadIdx.x]; }
    const float mean = ts * (1.f / 2048.f);
    mv[threadIdx.x] = mean;
    mv[64 + threadIdx.x] = tq * (1.f / 2048.f) - mean * mean;
  }
}

// featc = ((BN(x2) @ ew2^T + eb2) @ wg + bg) @ wl_w[:,64:]^T + wl_b
__global__ __launch_bounds__(256) void k_feat(
    const float* __restrict__ x2, const float* __restrict__ mv,
    const float* __restrict__ bnw, const float* __restrict__ bnb,
    const float* __restrict__ ew2, const float* __restrict__ eb2,
    const float* __restrict__ wg, const float* __restrict__ bg,
    const float* __restrict__ wl_w, const float* __restrict__ wl_b,
    float* __restrict__ featc) {
  extern __shared__ float smem[];
  float* sX = smem;          // 64x64
  float* sY = smem + 4096;   // 64x64
  float* sW = smem + 8192;   // 64x64 weight (reloaded per stage)
  const int tid = threadIdx.x, lane = tid & 31, wave = tid >> 5;
  const int ml = lane & 15, hi8 = (lane >> 4) << 3;
  const int n0 = blockIdx.x * 64;

  for (int i = tid; i < 64 * 64; i += 256) {
    const int r = i >> 6, c = i & 63;
    const float v = x2[(n0 + r) * H + c];
    sX[i] = (v - mv[c]) * rsqrtf(mv[64 + c] + BNEPS) * bnw[c] + bnb[c];
    sW[i] = ew2[c * H + r];  // sW[k][j] = ew2[j][k]
  }
  __syncthreads();
  for (int tt = wave; tt < 16; tt += 8) {  // feat -> sY
    const int m0 = (tt >> 2) << 4, c0 = (tt & 3) << 4;
    v8f acc;
    const float bias = eb2[c0 + ml];
    #pragma unroll
    for (int r = 0; r < 8; ++r) acc[r] = bias;
    acc = wmma_tile(&sX[m0 * 64], 64, &sW[c0], 64, 64, acc);
    #pragma unroll
    for (int r = 0; r < 8; ++r) sY[(m0 + r + hi8) * 64 + c0 + ml] = acc[r];
  }
  __syncthreads();
  for (int i = tid; i < 64 * 64; i += 256) sW[i] = wg[i];  // row-major direct
  __syncthreads();
  for (int tt = wave; tt < 16; tt += 8) {  // featg -> sX
    const int m0 = (tt >> 2) << 4, c0 = (tt & 3) << 4;
    v8f acc;
    const float bias = bg[0];
    #pragma unroll
    for (int r = 0; r < 8; ++r) acc[r] = bias;
    acc = wmma_tile(&sY[m0 * 64], 64, &sW[c0], 64, 64, acc);
    #pragma unroll
    for (int r = 0; r < 8; ++r) sX[(m0 + r + hi8) * 64 + c0 + ml] = acc[r];
  }
  __syncthreads();
  for (int i = tid; i < 64 * 64; i += 256)
    sW[i] = wl_w[(i & 63) * 128 + 64 + (i >> 6)];  // sW[k][j] = wl_w[j][64+k]
  __syncthreads();
  for (int tt = wave; tt < 16; tt += 8) {  // featc -> global
    const int m0 = (tt >> 2) << 4, c0 = (tt & 3) << 4;
    v8f acc;
    const float bias = wl_b[c0 + ml];
    #pragma unroll
    for (int r = 0; r < 8; ++r) acc[r] = bias;
    acc = wmma_tile(&sX[m0 * 64], 64, &sW[c0], 64, 64, acc);
    #pragma unroll
    for (int r = 0; r < 8; ++r)
      featc[(n0 + m0 + r + hi8) * H + c0 + ml] = acc[r];
  }
}

// ---------------------------------------------------------------------------
// Fused VGRU step: one workgroup = one batch b x 64-node chunk.
// RU source nodes for chunk [n0,n0+64): r-half = [n0/2, n0/2+32),
// u-half = [N/2+n0/2, N/2+n0/2+32).
// ---------------------------------------------------------------------------
__global__ __launch_bounds__(256) void k_step(
    const float* __restrict__ hs_prev, const float* __restrict__ inputs,
    const float* __restrict__ w1, const float* __restrict__ b1,
    const float* __restrict__ w2, const float* __restrict__ b2,
    const float* __restrict__ wl_w, const float* __restrict__ featc,
    float* __restrict__ hs_next, int t, int first) {
  extern __shared__ float smem[];
  float* RU = smem;          // 64 x 128  gate activations (src rows)
  float* W  = smem + 8192;   // 64 x 128  weights (W1', then W2|WL)
  float* HS = smem + 16384;  // 64 x 64   src hs, later tanh-c
  float* HD = smem + 20480;  // 64 x 64   dest hs
  float* T  = smem + 24576;  // 64 x 64   r*hs, later new
  float* X  = smem + 28672;  // 128       x_t at src / dest nodes
  const int tid = threadIdx.x, lane = tid & 31, wave = tid >> 5;
  const int ml = lane & 15, hi8 = (lane >> 4) << 3;
  const int b = blockIdx.x >> 5;
  const int n0 = (blockIdx.x & 31) << 6;
  const int s0 = n0 >> 1;
  const float* xt = inputs + (size_t)(b * SEQL + t) * N_NODES;

  // Phase 1: stage W1' (rows 1..64 of w1) and hs tiles
  for (int i = tid; i < 64 * 128; i += 256) W[i] = w1[128 + i];
  for (int i = tid; i < 64 * 64; i += 256) {
    const int r = i >> 6, c = i & 63;
    const int src = (r < 32) ? (s0 + r) : (N_NODES / 2 + s0 + (r - 32));
    HS[i] = first ? 0.f : hs_prev[(size_t)((b << 11) + src) * H + c];
    HD[i] = first ? 0.f : hs_prev[(size_t)((b << 11) + n0 + r) * H + c];
  }
  if (tid < 64) {
    const int src = (tid < 32) ? (s0 + tid) : (N_NODES / 2 + s0 + (tid - 32));
    X[tid] = xt[src];
    X[64 + tid] = xt[n0 + tid];
  }
  __syncthreads();

  // Phase 2: RU = sigmoid(x_src (x) w1[0] + HS @ W1' + b1)   (64 x 128)
  for (int tt = wave; tt < 32; tt += 8) {
    const int m0 = (tt >> 3) << 4, c0 = (tt & 7) << 4;
    v8f acc;
    const float bias = b1[c0 + ml], w0 = w1[c0 + ml];
    #pragma unroll
    for (int r = 0; r < 8; ++r) acc[r] = bias + X[m0 + r + hi8] * w0;
    acc = wmma_tile(&HS[m0 * 64], 64, &W[c0], 128, 64, acc);
    #pragma unroll
    for (int r = 0; r < 8; ++r)
      RU[(m0 + r + hi8) * 128 + c0 + ml] = 1.f / (1.f + __expf(-acc[r]));
  }
  __syncthreads();

  // Phase 3: reload weights (W2 | WL) side by side; T = r * HD
  for (int i = tid; i < 64 * 64; i += 256) {
    const int k = i >> 6, j = i & 63;
    W[k * 128 + j]      = w2[(k + 1) * 64 + j];   // W2'[k][j]
    W[k * 128 + 64 + j] = wl_w[j * 128 + k];      // WL[k][j] = wl_w[j][k]
  }
  for (int i = tid; i < 64 * 64; i += 256) {
    const int j = i >> 6, k = i & 63;
    T[i] = RU[(j >> 1) * 128 + ((j & 1) << 6) + k] * HD[i];
  }
  __syncthreads();

  // Phase 4: c = tanh(x_dst (x) w2[0] + T @ W2' + b2) -> HS (reuse)
  for (int tt = wave; tt < 16; tt += 8) {
    const int m0 = (tt >> 2) << 4, c0 = (tt & 3) << 4;
    v8f acc;
    const float bias = b2[c0 + ml], w0 = w2[c0 + ml];
    #pragma unroll
    for (int r = 0; r < 8; ++r) acc[r] = bias + X[64 + m0 + r + hi8] * w0;
    acc = wmma_tile(&T[m0 * 64], 64, &W[c0], 128, 64, acc);
    #pragma unroll
    for (int r = 0; r < 8; ++r)
      HS[(m0 + r + hi8) * 64 + c0 + ml] = tanhf(acc[r]);
  }
  __syncthreads();

  // Phase 5: new = u*HD + (1-u)*c  -> T
  for (int i = tid; i < 64 * 64; i += 256) {
    const int j = i >> 6, k = i & 63;
    const float u = RU[(32 + (j >> 1)) * 128 + ((j & 1) << 6) + k];
    T[i] = u * HD[i] + (1.f - u) * HS[i];
  }
  __syncthreads();

  // Phase 6: hs_next = new @ WL + featc (featc already holds wl_b + road term)
  for (int tt = wave; tt < 16; tt += 8) {
    const int m0 = (tt >> 2) << 4, c0 = (tt & 3) << 4;
    v8f acc;
    #pragma unroll
    for (int r = 0; r < 8; ++r)
      acc[r] = featc[(n0 + m0 + r + hi8) * H + c0 + ml];
    acc = wmma_tile(&T[m0 * 64], 64, &W[64 + c0], 128, 64, acc);
    #pragma unroll
    for (int r = 0; r < 8; ++r)
      hs_next[(size_t)((b << 11) + n0 + m0 + r + hi8) * H + c0 + ml] = acc[r];
  }
}

// ---------------------------------------------------------------------------

extern "C" void kernel_launch(void* const* d_in, const int* in_sizes, int n_in,
                              void* d_out, int out_size, void* d_ws, size_t ws_size,
                              hipStream_t stream) {
  (void)in_sizes; (void)n_in; (void)out_size; (void)ws_size;
  const float* h    = (const float*)d_in[0];
  const float* adj  = (const float*)d_in[1];
  const float* inp  = (const float*)d_in[2];
  const float* ew1  = (const float*)d_in[3];
  const float* eb1  = (const float*)d_in[4];
  const float* ew2  = (const float*)d_in[5];
  const float* eb2  = (const float*)d_in[6];
  const float* eps1 = (const float*)d_in[7];
  const float* bnw  = (const float*)d_in[8];
  const float* bnb  = (const float*)d_in[9];
  const float* w1   = (const float*)d_in[10];
  const float* b1   = (const float*)d_in[11];
  const float* w2   = (const float*)d_in[12];
  const float* b2   = (const float*)d_in[13];
  const float* wg   = (const float*)d_in[14];
  const float* bg   = (const float*)d_in[15];
  const float* wl_w = (const float*)d_in[16];
  const float* wl_b = (const float*)d_in[17];

  float* ws    = (float*)d_ws;
  float* hsA   = ws;                    // 8388608 floats (33.5 MB)
  float* xh    = hsA + 8388608;         // 131072
  float* x2    = xh + 131072;           // 131072
  float* featc = x2 + 131072;           // 131072
  float* deg   = featc + 131072;        // 2048
  float* mv    = deg + 2048;            // 128
  float* out   = (float*)d_out;

  k_xh  <<<32,  256, 65536, stream>>>(h, ew1, eb1, xh);
  k_deg <<<256, 256, 0,     stream>>>(adj, deg);
  k_pool<<<32,  256, 32768, stream>>>(adj, xh, deg, eps1, x2);
  k_stats<<<1,  256, 0,     stream>>>(x2, mv);
  k_feat<<<32,  256, 49152, stream>>>(x2, mv, bnw, bnb, ew2, eb2, wg, bg,
                                      wl_w, wl_b, featc);

  const float* prev = hsA;  // unused on first step (first=1 zeroes hs)
  for (int t = 0; t < SEQL; ++t) {
    float* next = (t & 1) ? out : hsA;  // t=11 (last) -> d_out
    k_step<<<2048, 256, 115200, stream>>>(prev, inp, w1, b1, w2, b2, wl_w,
                                          featc, next, t, t == 0 ? 1 : 0);
    prev = next;
  }
}